// TopKSparseAutoencoder_52012053954861
// MI455X (gfx1250) — compile-verified
//
#include <hip/hip_runtime.h>

typedef __attribute__((ext_vector_type(2))) float v2f;
typedef __attribute__((ext_vector_type(8))) float v8f;
typedef __attribute__((ext_vector_type(4))) int   v4i;

#define N_TOKENS 32768
#define D_IN     768
#define D_HID    3072
#define TOPK     64

#if __has_builtin(__builtin_amdgcn_global_load_async_to_lds_b128) && \
    __has_builtin(__builtin_amdgcn_s_wait_asynccnt)
#define USE_ASYNC 1
#else
#define USE_ASYNC 0
#endif

// ---------------------------------------------------------------------------
// Kernel 1: transpose W_dec (768 x 3072) -> WdT (3072 x 768) in workspace
// ---------------------------------------------------------------------------
__global__ __launch_bounds__(256) void k_transpose(const float* __restrict__ Wd,
                                                   float* __restrict__ WdT) {
    __shared__ float tile[32][33];
    int x = blockIdx.x * 32 + threadIdx.x;          // col of Wd (0..3071)
    int y0 = blockIdx.y * 32;                       // row of Wd
    #pragma unroll
    for (int i = threadIdx.y; i < 32; i += 8)
        tile[i][threadIdx.x] = Wd[(size_t)(y0 + i) * D_HID + x];
    __syncthreads();
    int xo = blockIdx.y * 32 + threadIdx.x;         // col of WdT (0..767)
    int yo0 = blockIdx.x * 32;                      // row of WdT
    #pragma unroll
    for (int i = threadIdx.y; i < 32; i += 8)
        WdT[(size_t)(yo0 + i) * D_IN + xo] = tile[threadIdx.x][i];
}

// ---------------------------------------------------------------------------
// Kernel 2: encoder GEMM  pre = z @ W_enc^T + b_enc   (WMMA f32 16x16x4)
// 256 threads = 8 waves; block computes 64(M) x 64(H); K double-buffered in
// 64x32 LDS chunks filled by GLOBAL_LOAD_ASYNC_TO_LDS_B128 when available.
// LDS row stride 36 floats: 16B-aligned fills, conflict-free b64 frag loads.
// ---------------------------------------------------------------------------
#define ASTR 36
#define KCHUNK 32
#define NCHUNK (D_IN / KCHUNK)   // 24

#if USE_ASYNC
__device__ __forceinline__ void async_b128(const float* g, float* l) {
    // builtin params are (int4 __device__*, int4 __shared__*, imm offset, imm cpol);
    // generic-typed pointers convert implicitly in HIP.
    __builtin_amdgcn_global_load_async_to_lds_b128((v4i*)g, (v4i*)l, 0, 0);
}
#endif

__device__ __forceinline__ v8f wmma_f32(v2f a, v2f b, v8f c) {
    return __builtin_amdgcn_wmma_f32_16x16x4_f32(false, a, false, b,
                                                 (short)0, c, false, false);
}

__device__ __forceinline__ void enc_compute_chunk(const float* AsB, const float* WsB,
                                                  int m_sub, int n_sub0, int row_in,
                                                  int kh, v8f& acc0, v8f& acc1) {
    #pragma unroll
    for (int k = 0; k < KCHUNK; k += 4) {
        int kk = k + 2 * kh;
        v2f a  = *(const v2f*)(&AsB[(m_sub * 16 + row_in) * ASTR + kk]);
        v2f b0 = *(const v2f*)(&WsB[( n_sub0      * 16 + row_in) * ASTR + kk]);
        v2f b1 = *(const v2f*)(&WsB[((n_sub0 + 1) * 16 + row_in) * ASTR + kk]);
        acc0 = wmma_f32(a, b0, acc0);
        acc1 = wmma_f32(a, b1, acc1);
    }
}

__global__ __launch_bounds__(256) void k_encoder(const float* __restrict__ z,
                                                 const float* __restrict__ W,
                                                 const float* __restrict__ b_enc,
                                                 float* __restrict__ pre) {
#if USE_ASYNC
    __shared__ float As[2][64 * ASTR];
    __shared__ float Ws[2][64 * ASTR];
#else
    __shared__ float As[1][64 * ASTR];
    __shared__ float Ws[1][64 * ASTR];
#endif

    const int tid  = threadIdx.x;
    const int lane = tid & 31;
    const int wave = tid >> 5;
    const int m_sub  = wave & 3;
    const int n_sub0 = (wave >> 2) * 2;
    const int row_in = lane & 15;
    const int kh     = lane >> 4;

    const int m0 = blockIdx.x * 64;
    const int h0 = blockIdx.y * 64;

    // per-thread staging slots: (row, row+32) x same 4-float column
    const int rA = tid >> 3;            // 0..31
    const int c4 = (tid & 7) * 4;       // 0,4,..,28

    v8f acc0 = {};
    v8f acc1 = {};

#if USE_ASYNC
    auto issue = [&](int buf, int k0) {
        async_b128(&z[(size_t)(m0 + rA)      * D_IN + k0 + c4], &As[buf][ rA       * ASTR + c4]);
        async_b128(&z[(size_t)(m0 + rA + 32) * D_IN + k0 + c4], &As[buf][(rA + 32) * ASTR + c4]);
        async_b128(&W[(size_t)(h0 + rA)      * D_IN + k0 + c4], &Ws[buf][ rA       * ASTR + c4]);
        async_b128(&W[(size_t)(h0 + rA + 32) * D_IN + k0 + c4], &Ws[buf][(rA + 32) * ASTR + c4]);
    };
    issue(0, 0);
    int buf = 0;
    for (int c = 0; c < NCHUNK; ++c) {
        __builtin_amdgcn_s_wait_asynccnt(0);
        __syncthreads();                              // chunk c resident for all waves
        if (c + 1 < NCHUNK) issue(buf ^ 1, (c + 1) * KCHUNK);
        enc_compute_chunk(As[buf], Ws[buf], m_sub, n_sub0, row_in, kh, acc0, acc1);
        buf ^= 1;
    }
#else
    for (int k0 = 0; k0 < D_IN; k0 += KCHUNK) {
        __syncthreads();
        float4 az0 = *(const float4*)(&z[(size_t)(m0 + rA)      * D_IN + k0 + c4]);
        float4 az1 = *(const float4*)(&z[(size_t)(m0 + rA + 32) * D_IN + k0 + c4]);
        float4 aw0 = *(const float4*)(&W[(size_t)(h0 + rA)      * D_IN + k0 + c4]);
        float4 aw1 = *(const float4*)(&W[(size_t)(h0 + rA + 32) * D_IN + k0 + c4]);
        *(float4*)(&As[0][ rA       * ASTR + c4]) = az0;
        *(float4*)(&As[0][(rA + 32) * ASTR + c4]) = az1;
        *(float4*)(&Ws[0][ rA       * ASTR + c4]) = aw0;
        *(float4*)(&Ws[0][(rA + 32) * ASTR + c4]) = aw1;
        __syncthreads();
        enc_compute_chunk(As[0], Ws[0], m_sub, n_sub0, row_in, kh, acc0, acc1);
    }
#endif

    const int gc0 = h0 + n_sub0 * 16 + row_in;
    const float bias0 = b_enc[gc0];
    const float bias1 = b_enc[gc0 + 16];
    #pragma unroll
    for (int v = 0; v < 8; ++v) {
        int gr = m0 + m_sub * 16 + v + 8 * kh;        // C/D layout: VGPR v = row v / v+8
        pre[(size_t)gr * D_HID + gc0]      = acc0[v] + bias0;
        pre[(size_t)gr * D_HID + gc0 + 16] = acc1[v] + bias1;
    }
}

// ---------------------------------------------------------------------------
// Kernel 3: per-row top-64 radix select (deterministic), writes s + indices.
// 128 threads = 4 waves, one row per wave.
// ---------------------------------------------------------------------------
__device__ __forceinline__ unsigned order_key(float v) {
    unsigned u = __float_as_uint(v);
    return u ^ ((u >> 31) ? 0xFFFFFFFFu : 0x80000000u);
}

__global__ __launch_bounds__(128) void k_topk(const float* __restrict__ pre,
                                              float* __restrict__ s_out,
                                              int* __restrict__ idx_out) {
    __shared__ float    rowbuf[4][D_HID];
    __shared__ unsigned hist[4][256];
    __shared__ unsigned binb_sh[4];
    __shared__ unsigned rem_sh[4];

    const int tid  = threadIdx.x;
    const int lane = tid & 31;
    const int wave = tid >> 5;
    const int row  = blockIdx.x * 4 + wave;
    const float* prow = pre + (size_t)row * D_HID;

    #pragma unroll
    for (int i = lane; i < 256; i += 32) hist[wave][i] = 0u;
    __syncthreads();

    for (int t = 0; t < D_HID / 32; ++t) {
        int j = t * 32 + lane;
        float v = prow[j];
        rowbuf[wave][j] = v;
        atomicAdd(&hist[wave][order_key(v) >> 24], 1u);
    }
    __syncthreads();

    if (lane == 0) {
        unsigned above = 0, bb = 0, r = TOPK;
        for (int b = 255; b >= 0; --b) {
            unsigned c = hist[wave][b];
            if (above + c >= (unsigned)TOPK) { bb = (unsigned)b; r = TOPK - above; break; }
            above += c;
        }
        binb_sh[wave] = bb;
        rem_sh[wave]  = r;
    }
    __syncthreads();

    const unsigned binb = binb_sh[wave];
    const unsigned rem  = rem_sh[wave];
    const unsigned ltm  = (1u << lane) - 1u;

    unsigned cntB = 0, cntS = 0;
    for (int t = 0; t < D_HID / 32; ++t) {
        int j = t * 32 + lane;
        float v = rowbuf[wave][j];
        unsigned bin = order_key(v) >> 24;
        bool isB = (bin == binb);
        unsigned mB = __builtin_amdgcn_ballot_w32(isB);
        unsigned rankB = cntB + __popc(mB & ltm);
        bool sel = (bin > binb) | (isB & (rankB < rem));
        unsigned mS = __builtin_amdgcn_ballot_w32(sel);
        unsigned pos = cntS + __popc(mS & ltm);
        if (sel) idx_out[(size_t)row * TOPK + pos] = j;
        s_out[(size_t)row * D_HID + j] = sel ? v : 0.0f;
        cntB += __popc(mB);
        cntS += __popc(mS);
    }
}

// ---------------------------------------------------------------------------
// Kernel 4: sparse decoder  z_hat = sum_k val_k * WdT[idx_k,:] + b_dec
// 192 threads x float4 (768 = 192*4); WdT (9.4MB) is L2-resident.
// ---------------------------------------------------------------------------
__global__ __launch_bounds__(192) void k_decoder(const float* __restrict__ s,
                                                 const int* __restrict__ topk,
                                                 const float* __restrict__ WdT,
                                                 const float* __restrict__ b_dec,
                                                 float* __restrict__ zh) {
    __shared__ int   sidx[TOPK];
    __shared__ float sval[TOPK];
    const int row = blockIdx.x;
    const int tid = threadIdx.x;
    if (tid < TOPK) {
        int id = topk[(size_t)row * TOPK + tid];
        sidx[tid] = id;
        sval[tid] = s[(size_t)row * D_HID + id];
    }
    __syncthreads();

    float4 acc = make_float4(0.f, 0.f, 0.f, 0.f);
    #pragma unroll 8
    for (int i = 0; i < TOPK; ++i) {
        float v = sval[i];
        float4 w = *(const float4*)(WdT + (size_t)sidx[i] * D_IN + tid * 4);
        acc.x = fmaf(v, w.x, acc.x);
        acc.y = fmaf(v, w.y, acc.y);
        acc.z = fmaf(v, w.z, acc.z);
        acc.w = fmaf(v, w.w, acc.w);
    }
    float4 b = *(const float4*)(b_dec + tid * 4);
    acc.x += b.x; acc.y += b.y; acc.z += b.z; acc.w += b.w;
    *(float4*)(zh + (size_t)row * D_IN + tid * 4) = acc;
}

// ---------------------------------------------------------------------------
extern "C" void kernel_launch(void* const* d_in, const int* in_sizes, int n_in,
                              void* d_out, int out_size, void* d_ws, size_t ws_size,
                              hipStream_t stream) {
    const float* z     = (const float*)d_in[0];
    const float* W_enc = (const float*)d_in[1];
    const float* b_enc = (const float*)d_in[2];
    const float* W_dec = (const float*)d_in[3];
    const float* b_dec = (const float*)d_in[4];

    float* out = (float*)d_out;
    // output layout: z_hat | s | topk_idx | pre_acts  (reference return order)
    float* zhat = out;
    float* s    = out + (size_t)N_TOKENS * D_IN;
    int*   idx  = (int*)(out + (size_t)N_TOKENS * (D_IN + D_HID));
    float* pre  = out + (size_t)N_TOKENS * (D_IN + D_HID) + (size_t)N_TOKENS * TOPK;

    float* WdT = (float*)d_ws;   // 3072 x 768 f32 = 9.4 MB

    k_transpose<<<dim3(D_HID / 32, D_IN / 32), dim3(32, 8), 0, stream>>>(W_dec, WdT);
    k_encoder<<<dim3(N_TOKENS / 64, D_HID / 64), 256, 0, stream>>>(z, W_enc, b_enc, pre);
    k_topk<<<N_TOKENS / 4, 128, 0, stream>>>(pre, s, idx);
    k_decoder<<<N_TOKENS, 192, 0, stream>>>(s, idx, WdT, b_dec, zhat);
}